// Model_8718783611250
// MI455X (gfx1250) — compile-verified
//
#include <hip/hip_runtime.h>
#include <hip/hip_bf16.h>
#include <stdint.h>

// ---------------------------------------------------------------------------
// Hyperbolic RGAT (2 hops) + 2-layer COO GCN for MI455X (gfx1250, wave32).
//
// Memory-bound gather/scatter workload (~3.3 GB traffic -> ~140us @ 23.3TB/s).
// CDNA5-specific paths used:
//   * V_WMMA_F32_16X16X4_F32: each wave batches 5 edges; the 15 gathered
//     64-dim vectors form a 16x64 matrix M (1 pad row); 16 chained WMMAs
//     compute the Gram G = M*M^T, which supplies ALL dot products the
//     Mobius/exp/log-map algebra needs (everything is a linear combination
//     of {he,te,re} with Gram-derived coefficients).
//   * global_load_async_to_lds_b128/b64 + s_wait_asynccnt: async staging of
//     gathered rows into LDS (ASYNCcnt path).
//   * non-returning global_atomic_add_f32 for all scatter-sums.
//
// Workspace (floats): entb[12.8M] | sums[12.8M] | resb[12.8M] | cnt[200K]
//   = 154.4 MB. GCN ping-pong buffers alias entb/sums.
// ---------------------------------------------------------------------------

#define N_USER 100000
#define N_ITEM 50000
#define N_ENT  200000
#define D      64
#define N_EDGE 500000
#define NNZ    2000000
#define NUI    (N_USER + N_ITEM)          // 150000
#define EMB_N  ((size_t)NUI * D)          // 9,600,000
#define ENT_N  ((size_t)N_ENT * D)        // 12,800,000
#define ROWP   68                         // padded LDS row (floats), no bank conflicts

typedef __attribute__((ext_vector_type(2))) float v2f;
typedef __attribute__((ext_vector_type(8))) float v8f;

// ---------------- wave32 reduction helper (finalize kernel) ----------------
__device__ __forceinline__ float wave_sum(float v) {
  v += __shfl_xor(v, 16, 32);
  v += __shfl_xor(v, 8, 32);
  v += __shfl_xor(v, 4, 32);
  v += __shfl_xor(v, 2, 32);
  v += __shfl_xor(v, 1, 32);
  return v;
}
__device__ __forceinline__ float dot2w(float2 a, float2 b) {
  return wave_sum(a.x * b.x + a.y * b.y);
}

// ---------------- CDNA5 async global->LDS copies ----------------
__device__ __forceinline__ void async_b64_to_lds(unsigned lds_off, const void* g) {
  asm volatile("global_load_async_to_lds_b64 %0, %1, off"
               :: "v"(lds_off), "v"((unsigned long long)(uintptr_t)g) : "memory");
}
__device__ __forceinline__ void async_b128_to_lds(unsigned lds_off, const void* g) {
  asm volatile("global_load_async_to_lds_b128 %0, %1, off"
               :: "v"(lds_off), "v"((unsigned long long)(uintptr_t)g) : "memory");
}
__device__ __forceinline__ void wait_asynccnt0() {
#if defined(__has_builtin)
#if __has_builtin(__builtin_amdgcn_s_wait_asynccnt)
  __builtin_amdgcn_s_wait_asynccnt(0);
#else
  asm volatile("s_wait_asynccnt 0" ::: "memory");
#endif
#else
  asm volatile("s_wait_asynccnt 0" ::: "memory");
#endif
}

// ---------------- WMMA C/D layout accessors ----------------
// D[m][n]: vgpr (m&7), lane n + 16*(m>>3)   (16x16 f32 C/D layout, ISA 7.12.2)
__device__ __forceinline__ float v8_get(v8f v, int i) {
  switch (i & 7) {
    case 0: return v[0]; case 1: return v[1]; case 2: return v[2]; case 3: return v[3];
    case 4: return v[4]; case 5: return v[5]; case 6: return v[6]; default: return v[7];
  }
}
__device__ __forceinline__ float gram_at(v8f acc, int m, int n) {
  return __shfl(v8_get(acc, m & 7), n + ((m >> 3) << 4), 32);
}

// ---------------- RGAT edge kernel: one wave per 5 edges, WMMA Gram -------
__global__ __launch_bounds__(256) void rgat_edge_kernel(
    const float* __restrict__ ent, const float* __restrict__ rel,
    const int* __restrict__ head, const int* __restrict__ tail,
    const int* __restrict__ etype, float* __restrict__ sums,
    float* __restrict__ cnt) {
  __shared__ float smem[8 * 16 * ROWP];           // 8 waves x 16 rows x 68 floats
  const int lane = threadIdx.x & 31;
  const int wid = threadIdx.x >> 5;
  const int wave = blockIdx.x * 8 + wid;          // 100,000 waves exactly
  const int e0 = wave * 5;                        // first of 5 edges
  float* slot = smem + wid * 16 * ROWP;

  // ---- stage rows 0..14 (he,te,re per edge; row 15 = dummy) via async b128
  const int hlf = lane >> 4;                      // 0 or 1: which row of pair
  const int l16 = lane & 15;                      // 16B chunk id within row
  #pragma unroll
  for (int i = 0; i < 8; ++i) {
    int r = 2 * i + hlf;                          // row 0..15
    int e = (r >= 15) ? 4 : ((r * 11) >> 5);      // r/3 (row15 -> dup edge 4)
    int kind = (r >= 15) ? 2 : (r - 3 * e);       // 0=head,1=tail,2=rel
    int idx;
    const float* base;
    if (kind == 0)      { idx = head[e0 + e];      base = ent; }
    else if (kind == 1) { idx = tail[e0 + e];      base = ent; }
    else                { idx = etype[e0 + e] - 1; base = rel; }
    const float* src = base + (size_t)idx * D + 4 * l16;
    unsigned lds = (unsigned)(uintptr_t)(slot + r * ROWP + 4 * l16);
    async_b128_to_lds(lds, src);
  }
  wait_asynccnt0();

  // ---- Gram = M * M^T via 16 chained V_WMMA_F32_16X16X4_F32 --------------
  // A fragment step k: lane L holds M[L&15][4k + 2*(L>>4) + {0,1}].
  // For symmetric Gram the B (=M^T) fragment layout coincides with A.
  v8f acc = {};
  const int frow = lane & 15;
  const int fcol = (lane >> 4) * 2;
  #pragma unroll
  for (int k = 0; k < 16; ++k) {
    float2 f = *(const float2*)(slot + frow * ROWP + 4 * k + fcol);
    v2f afrag = {f.x, f.y};
    acc = __builtin_amdgcn_wmma_f32_16x16x4_f32(
        false, afrag, false, afrag, (short)0, acc, false, false);
  }

  // ---- per-edge scalar pipeline (wave-uniform) + output ------------------
  #pragma unroll
  for (int e = 0; e < 5; ++e) {
    const int ra = 3 * e, rb = 3 * e + 1, rc = 3 * e + 2;
    float a = gram_at(acc, ra, ra);   // ||he||^2
    float b = gram_at(acc, rb, rb);   // ||te||^2
    float c = gram_at(acc, rc, rc);   // ||re||^2
    float p = gram_at(acc, ra, rb);   // <he,te>
    float q = gram_at(acc, ra, rc);   // <he,re>
    float w = gram_at(acc, rb, rc);   // <te,re>

    // ricci = (te+re)/max(||te+re||,1e-12)
    float rin = 1.0f / fmaxf(sqrtf(fmaxf(b + 2.0f * w + c, 0.0f)), 1e-12f);
    // hh = expmap0(he) = sc*he
    float nhe = fmaxf(sqrtf(fmaxf(a, 0.0f)), 1e-15f);
    float sc = tanhf(nhe) / nhe;
    float hh2 = sc * sc * a;
    float lam = 2.0f / fmaxf(1.0f - hh2, 1e-15f);
    // ht = mobius(hh, a1*te) = h1*he + t1*te
    float nte = fmaxf(sqrtf(fmaxf(b, 0.0f)), 1e-15f);
    float a1 = tanhf(0.5f * lam * nte) / nte;
    float y2 = a1 * a1 * b, xy = sc * a1 * p;
    float Am = 1.0f + 2.0f * xy + y2, Bm = 1.0f - hh2;
    float inv = 1.0f / fmaxf(1.0f + 2.0f * xy + hh2 * y2, 1e-15f);
    float h1 = Am * sc * inv, t1 = Bm * a1 * inv;
    // hr = mobius(hh, a2*re) = h2*he + r2*re
    float nre = fmaxf(sqrtf(fmaxf(c, 0.0f)), 1e-15f);
    float a2 = tanhf(0.5f * lam * nre) / nre;
    y2 = a2 * a2 * c; xy = sc * a2 * q;
    Am = 1.0f + 2.0f * xy + y2;
    inv = 1.0f / fmaxf(1.0f + 2.0f * xy + hh2 * y2, 1e-15f);
    float h2 = Am * sc * inv, r2 = Bm * a2 * inv;
    // r0 = mobius(ht, hr) = ch*he + ct*te + cr*re
    float ht2 = h1 * h1 * a + 2.0f * h1 * t1 * p + t1 * t1 * b;
    float hr2 = h2 * h2 * a + 2.0f * h2 * r2 * q + r2 * r2 * c;
    float hth = h1 * h2 * a + h1 * r2 * q + t1 * h2 * p + t1 * r2 * w;
    Am = 1.0f + 2.0f * hth + hr2; Bm = 1.0f - ht2;
    inv = 1.0f / fmaxf(1.0f + 2.0f * hth + ht2 * hr2, 1e-15f);
    float ch = (Am * h1 + Bm * h2) * inv;
    float ct = Am * t1 * inv;
    float cr = Bm * r2 * inv;
    // project to ball
    float n02 = fmaxf(ch * ch * a + ct * ct * b + cr * cr * c +
                      2.0f * (ch * ct * p + ch * cr * q + ct * cr * w), 0.0f);
    float n0 = fmaxf(sqrtf(n02), 1e-15f);
    if (n0 > 0.999f) {
      float qq = 0.999f / n0;
      ch *= qq; ct *= qq; cr *= qq; n02 *= qq * qq;
    }
    // sub = mobius(-hh, r0) = sh*he + st*te + sr*re
    xy = -sc * (ch * a + ct * p + cr * q);
    Am = 1.0f + 2.0f * xy + n02; Bm = 1.0f - hh2;
    inv = 1.0f / fmaxf(1.0f + 2.0f * xy + hh2 * n02, 1e-15f);
    float sh = (Am * (-sc) + Bm * ch) * inv;
    float st = Bm * ct * inv;
    float sr = Bm * cr * inv;
    float ns2 = fmaxf(sh * sh * a + st * st * b + sr * sr * c +
                      2.0f * (sh * st * p + sh * sr * q + st * sr * w), 0.0f);
    float nsub = fmaxf(sqrtf(ns2), 1e-15f);
    float g = (2.0f / lam) * atanhf(fminf(nsub, 1.0f - 1e-5f)) / nsub;
    float alpha = g * sh;
    float beta  = g * st + 1e-7f * rin;   // ricci folds into te,re coeffs
    float gamma = g * sr + 1e-7f * rin;

    // out = relu(alpha*he + beta*te + gamma*re); scatter-add by head
    int h = head[e0 + e];
    float2 hev = *(const float2*)(slot + ra * ROWP + 2 * lane);
    float2 tev = *(const float2*)(slot + rb * ROWP + 2 * lane);
    float2 rev = *(const float2*)(slot + rc * ROWP + 2 * lane);
    float ox = fmaxf(alpha * hev.x + beta * tev.x + gamma * rev.x, 0.0f);
    float oy = fmaxf(alpha * hev.y + beta * tev.y + gamma * rev.y, 0.0f);
    float* dst = sums + (size_t)h * D + 2 * lane;
    atomicAdd(dst, ox);
    atomicAdd(dst + 1, oy);
    if (lane == 0) atomicAdd(cnt + h, 1.0f);
  }
}

// ---------------- mean + l2norm + residual: one wave per entity ----------
__global__ __launch_bounds__(256) void rgat_finalize_kernel(
    const float* __restrict__ sums, const float* __restrict__ cnt,
    float* __restrict__ ent_out, float* __restrict__ res) {
  int lane = threadIdx.x & 31;
  int e = blockIdx.x * 8 + (threadIdx.x >> 5);   // 200,000 exactly
  float c = fmaxf(cnt[e], 1.0f);
  size_t o = (size_t)e * D + 2 * lane;
  float2 v = *(const float2*)(sums + o);
  v.x /= c; v.y /= c;
  float inv = 1.0f / fmaxf(sqrtf(dot2w(v, v)), 1e-12f);
  v.x *= inv; v.y *= inv;
  *(float2*)(ent_out + o) = v;
  float2 rr = *(const float2*)(res + o);
  rr.x = 0.5f * rr.x + v.x;
  rr.y = 0.5f * rr.y + v.y;
  *(float2*)(res + o) = rr;
}

// ---------------- embeds = concat(u, ent_res[:N_ITEM]); init total ------
__global__ __launch_bounds__(256) void build_embeds_kernel(
    const float* __restrict__ u, const float* __restrict__ res,
    float* __restrict__ cur, float* __restrict__ out) {
  size_t i = (size_t)blockIdx.x * 256 + threadIdx.x;
  float v = (i < (size_t)N_USER * D) ? u[i] : res[i - (size_t)N_USER * D];
  cur[i] = v;
  out[i] = v;
}

// ---------------- COO SpMM: one wave per nnz, async-LDS staged gather ----
__global__ __launch_bounds__(256) void spmm_kernel(
    const float* __restrict__ cur, const float* __restrict__ val,
    const int* __restrict__ rows, const int* __restrict__ cols,
    float* __restrict__ out) {
  __shared__ float smem[8 * D];
  int lane = threadIdx.x & 31;
  int wid = threadIdx.x >> 5;
  int e = blockIdx.x * 8 + wid;                  // 2,000,000 exactly
  int c = cols[e];
  int r = rows[e];
  float v = val[e];
  const float* src = cur + (size_t)c * D + 2 * lane;
  float* slotp = &smem[wid * D + 2 * lane];
  async_b64_to_lds((unsigned)(uintptr_t)slotp, src);
  wait_asynccnt0();
  float2 x = *(const float2*)slotp;
  float* dst = out + (size_t)r * D + 2 * lane;
  atomicAdd(dst, v * x.x);
  atomicAdd(dst + 1, v * x.y);
}

// ---------------- total += layer ----------------
__global__ __launch_bounds__(256) void add_kernel(
    float* __restrict__ out, const float* __restrict__ x) {
  size_t i = (size_t)blockIdx.x * 256 + threadIdx.x;
  out[i] += x[i];
}

// ---------------------------------------------------------------------------
extern "C" void kernel_launch(void* const* d_in, const int* in_sizes, int n_in,
                              void* d_out, int out_size, void* d_ws, size_t ws_size,
                              hipStream_t stream) {
  const float* uEmb   = (const float*)d_in[0];
  const float* eEmb   = (const float*)d_in[1];
  const float* rEmb   = (const float*)d_in[2];
  const float* adjval = (const float*)d_in[3];
  const int*   ehead  = (const int*)d_in[4];
  const int*   etail  = (const int*)d_in[5];
  const int*   etype  = (const int*)d_in[6];
  const int*   arow   = (const int*)d_in[7];
  const int*   acol   = (const int*)d_in[8];
  float* out = (float*)d_out;

  float* ws   = (float*)d_ws;
  float* entb = ws;                       // 12.8M floats (also GCN bufA)
  float* sums = ws + ENT_N;               // 12.8M floats (also GCN bufB)
  float* resb = ws + 2 * ENT_N;           // 12.8M floats (ent_res)
  float* cntb = ws + 3 * ENT_N;           // 200K floats

  hipMemcpyAsync(resb, eEmb, ENT_N * sizeof(float),
                 hipMemcpyDeviceToDevice, stream);

  // ---- 2 RGAT hops ----
  for (int hop = 0; hop < 2; ++hop) {
    hipMemsetAsync(sums, 0, ENT_N * sizeof(float), stream);
    hipMemsetAsync(cntb, 0, (size_t)N_ENT * sizeof(float), stream);
    const float* src = (hop == 0) ? eEmb : entb;
    rgat_edge_kernel<<<N_EDGE / 5 / 8, 256, 0, stream>>>(src, rEmb, ehead,
                                                         etail, etype, sums, cntb);
    rgat_finalize_kernel<<<N_ENT / 8, 256, 0, stream>>>(sums, cntb, entb, resb);
  }

  // ---- build embeds: cur0 = total = concat(u, ent_res[:N_ITEM]) ----
  int ew_blocks = (int)(EMB_N / 256);     // 37500 exactly
  build_embeds_kernel<<<ew_blocks, 256, 0, stream>>>(uEmb, resb, entb, out);

  // ---- GCN layer 1: next(sums) = A * cur(entb); total += next ----
  hipMemsetAsync(sums, 0, EMB_N * sizeof(float), stream);
  spmm_kernel<<<NNZ / 8, 256, 0, stream>>>(entb, adjval, arow, acol, sums);
  add_kernel<<<ew_blocks, 256, 0, stream>>>(out, sums);

  // ---- GCN layer 2: next(entb) = A * cur(sums); total += next ----
  hipMemsetAsync(entb, 0, EMB_N * sizeof(float), stream);
  spmm_kernel<<<NNZ / 8, 256, 0, stream>>>(sums, adjval, arow, acol, entb);
  add_kernel<<<ew_blocks, 256, 0, stream>>>(out, entb);
}